// GraphSAGE_10204842295688
// MI455X (gfx1250) — compile-verified
//
#include <hip/hip_runtime.h>

typedef float v2f __attribute__((ext_vector_type(2)));
typedef float v8f __attribute__((ext_vector_type(8)));

#define DFEAT 256
#define APAD  260   // LDS row stride (floats): 260 % 64 == 4 -> conflict-free b64 A reads

// ---------------- utility: zero fill (ws is poisoned, must zero agg/deg) ----
__global__ void sage_zero_f32(float* __restrict__ p, long n) {
    long i = (long)blockIdx.x * blockDim.x + threadIdx.x;
    if (i < n) p[i] = 0.0f;
}

// ---------------- in-degree count -------------------------------------------
__global__ void sage_deg_count(const int* __restrict__ dst, float* __restrict__ deg, int E) {
    int e = blockIdx.x * blockDim.x + threadIdx.x;
    if (e < E) atomicAdd(&deg[dst[e]], 1.0f);
}

// in-place deg -> inv_deg  (zero-degree nodes -> 0)
__global__ void sage_inv_deg(float* __restrict__ deg, int N) {
    int i = blockIdx.x * blockDim.x + threadIdx.x;
    if (i < N) {
        float d = deg[i];
        deg[i] = (d > 0.0f) ? (1.0f / d) : 0.0f;
    }
}

// ---------------- scatter-add of source features into agg -------------------
// one thread per (edge, 16B feature chunk): 64 chunks cover D=256 floats.
__global__ void sage_scatter_add(const float* __restrict__ h,
                                 const int* __restrict__ src,
                                 const int* __restrict__ dst,
                                 float* __restrict__ agg, int E) {
    long idx = (long)blockIdx.x * blockDim.x + threadIdx.x;
    long total = (long)E * 64;
    if (idx >= total) return;
    int e = (int)(idx >> 6);
    int c = (int)(idx & 63);
    int s = src[e];
    int d = dst[e];
    const float4 v = ((const float4*)(h + (long)s * DFEAT))[c];
    float* p = agg + (long)d * DFEAT + (long)c * 4;
    atomicAdd(p + 0, v.x);
    atomicAdd(p + 1, v.y);
    atomicAdd(p + 2, v.z);
    atomicAdd(p + 3, v.w);
}

// ---------------- fused SAGE layer GEMM via V_WMMA_F32_16X16X4_F32 ----------
// out = relu( h @ Ws + (agg * inv_deg) @ Wn + b )
// block = 512 threads = 16 waves; wave w owns output tile [row0:row0+16, 16w:16w+16].
// A tiles (h rows + inv_deg-scaled agg rows) are shared by all 16 waves ->
// staged once in LDS (conflict-free padded layout), W streams from L2.
// N is a multiple of 16 (50000 = 3125*16): EXEC is all-1s at every WMMA.
__global__ __launch_bounds__(512) void sage_gemm_wmma(
    const float* __restrict__ h, const float* __restrict__ agg,
    const float* __restrict__ invd,
    const float* __restrict__ Ws, const float* __restrict__ Wn,
    const float* __restrict__ bias, float* __restrict__ out) {

    __shared__ float shA[16][APAD];   // h tile      [16 rows][256 K]
    __shared__ float shG[16][APAD];   // scaled agg  [16 rows][256 K]

    const int tid  = threadIdx.x;
    const int lane = tid & 31;
    const int wave = tid >> 5;        // 0..15
    const int row0 = blockIdx.x * 16;

    // ---- stage A tiles: thread (wave=row, lane) loads 8 consecutive floats ----
    {
        const int r = wave;                 // 0..15
        const int c = lane * 8;             // 0..248, wave covers a full 1KB row
        const float inv = invd[row0 + r];
        const float4* hp = (const float4*)(h   + (size_t)(row0 + r) * DFEAT + c);
        const float4* gp = (const float4*)(agg + (size_t)(row0 + r) * DFEAT + c);
        float4 h0 = hp[0], h1 = hp[1];
        float4 g0 = gp[0], g1 = gp[1];
        g0.x *= inv; g0.y *= inv; g0.z *= inv; g0.w *= inv;
        g1.x *= inv; g1.y *= inv; g1.z *= inv; g1.w *= inv;
        *(float4*)&shA[r][c]     = h0;
        *(float4*)&shA[r][c + 4] = h1;
        *(float4*)&shG[r][c]     = g0;
        *(float4*)&shG[r][c + 4] = g1;
    }
    __syncthreads();

    // ---- per-wave 16x16 tile compute ----
    const int half = lane >> 4;            // K half-select (A/B layout)
    const int n    = lane & 15;            // M row (A) / N col (B,C,D)
    const int col  = wave * 16 + n;

    // B operands: lane holds K = k0 + 2*half + {0,1} of column `col`
    const float* wsB = Ws + (size_t)(2 * half) * DFEAT + col;
    const float* wnB = Wn + (size_t)(2 * half) * DFEAT + col;

    // C/D layout: VGPR r -> M = r + 8*half, N = n  => bias depends only on col
    v8f acc;
    const float bv = bias[col];
#pragma unroll
    for (int r = 0; r < 8; ++r) acc[r] = bv;

#pragma unroll 4
    for (int k0 = 0; k0 < DFEAT; k0 += 4) {
        // A fragments from LDS (bank-conflict-free b64 per lane)
        v2f a = *(const v2f*)&shA[n][k0 + 2 * half];
        v2f g = *(const v2f*)&shG[n][k0 + 2 * half];

        // self path: h @ Ws
        v2f w;
        w.x = wsB[(size_t)k0 * DFEAT];
        w.y = wsB[(size_t)k0 * DFEAT + DFEAT];
        acc = __builtin_amdgcn_wmma_f32_16x16x4_f32(
            false, a, false, w, (short)0, acc, false, false);

        // neighbor path: (agg * inv_deg) @ Wn
        v2f w2;
        w2.x = wnB[(size_t)k0 * DFEAT];
        w2.y = wnB[(size_t)k0 * DFEAT + DFEAT];
        acc = __builtin_amdgcn_wmma_f32_16x16x4_f32(
            false, g, false, w2, (short)0, acc, false, false);
    }

#pragma unroll
    for (int r = 0; r < 8; ++r) {
        int orow = row0 + r + 8 * half;
        out[(size_t)orow * DFEAT + col] = fmaxf(acc[r], 0.0f);
    }
}

// ---------------------------------------------------------------------------
extern "C" void kernel_launch(void* const* d_in, const int* in_sizes, int n_in,
                              void* d_out, int out_size, void* d_ws, size_t ws_size,
                              hipStream_t stream) {
    const float* x   = (const float*)d_in[0];
    const int*   src = (const int*)d_in[1];
    const int*   dst = (const int*)d_in[2];
    const float* Ws0 = (const float*)d_in[3];
    const float* Wn0 = (const float*)d_in[4];
    const float* b0  = (const float*)d_in[5];
    const float* Ws1 = (const float*)d_in[6];
    const float* Wn1 = (const float*)d_in[7];
    const float* b1  = (const float*)d_in[8];
    float* out = (float*)d_out;

    const int N = in_sizes[0] / DFEAT;   // 50000
    const int E = in_sizes[1];           // 800000

    // workspace layout: h1[N*D] | agg[N*D] | deg[N]
    float* h1  = (float*)d_ws;
    float* agg = h1 + (size_t)N * DFEAT;
    float* deg = agg + (size_t)N * DFEAT;

    const long nd_feat = (long)N * DFEAT;
    const long scatter_threads = (long)E * 64;

    // in-degree -> inv_deg (in place)
    sage_zero_f32<<<(unsigned)((N + 255) / 256), 256, 0, stream>>>(deg, N);
    sage_deg_count<<<(unsigned)((E + 255) / 256), 256, 0, stream>>>(dst, deg, E);
    sage_inv_deg<<<(unsigned)((N + 255) / 256), 256, 0, stream>>>(deg, N);

    // ---- layer 0: x -> h1 ----
    sage_zero_f32<<<(unsigned)((nd_feat + 255) / 256), 256, 0, stream>>>(agg, nd_feat);
    sage_scatter_add<<<(unsigned)((scatter_threads + 255) / 256), 256, 0, stream>>>(
        x, src, dst, agg, E);
    sage_gemm_wmma<<<(unsigned)(N / 16), 512, 0, stream>>>(
        x, agg, deg, Ws0, Wn0, b0, h1);

    // ---- layer 1: h1 -> out ----
    sage_zero_f32<<<(unsigned)((nd_feat + 255) / 256), 256, 0, stream>>>(agg, nd_feat);
    sage_scatter_add<<<(unsigned)((scatter_threads + 255) / 256), 256, 0, stream>>>(
        h1, src, dst, agg, E);
    sage_gemm_wmma<<<(unsigned)(N / 16), 512, 0, stream>>>(
        h1, agg, deg, Ws1, Wn1, b1, out);
}